// MultiHeadGATLayer_24558622998849
// MI455X (gfx1250) — compile-verified
//
#include <hip/hip_runtime.h>
#include <hip/hip_bf16.h>
#include <stdint.h>

#define N_NODES   50000
#define E_EDGES   800000
#define H_HEADS   4
#define O_DIM     64
#define DIN_DIM   128
#define HO        256          // H_HEADS * O_DIM
#define NEG_SLOPE 0.2f
#define ENC_NEG_INF 0x007FFFFFu   // ordered-uint encoding of -inf

typedef float v2f __attribute__((ext_vector_type(2)));
typedef float v8f __attribute__((ext_vector_type(8)));

// ---- monotone float <-> uint encoding so atomicMax(uint) == float max ----
__device__ __forceinline__ unsigned encOrdered(float f) {
    unsigned b = __float_as_uint(f);
    return (b & 0x80000000u) ? ~b : (b | 0x80000000u);
}
__device__ __forceinline__ float decOrdered(unsigned u) {
    unsigned b = (u & 0x80000000u) ? (u ^ 0x80000000u) : ~u;
    float f = __uint_as_float(b);
    return isfinite(f) ? f : 0.0f;   // matches reference's isfinite(m) fixup
}

__device__ __forceinline__ float leaky(float v) {
    return v > 0.0f ? v : v * NEG_SLOPE;
}

// -------------------- init: zero out, zero denom, m = -inf --------------------
__global__ void k_init(float* __restrict__ out, float* __restrict__ denom,
                       unsigned* __restrict__ m) {
    int i = blockIdx.x * blockDim.x + threadIdx.x;
    if (i < N_NODES * HO) out[i] = 0.0f;
    if (i < N_NODES * H_HEADS) { denom[i] = 0.0f; m[i] = ENC_NEG_INF; }
}

// -------------------- GEMM: Z[N][256] = X[N][128] * W[128][256] ---------------
// One wave per 16x16 output tile; K swept in steps of 4 via V_WMMA_F32_16X16X4_F32.
// A (16x4 f32):  lane%16 = M row; lane/16 selects K pair {0,1} / {2,3}.
// B (4x16 f32):  lane%16 = N col; lane/16 selects K pair (mirrors A).
// C/D (16x16):   VGPR r -> M = r + (lane/16)*8, N = lane%16.
__global__ void k_gemm_wmma(const float* __restrict__ X,
                            const float* __restrict__ W,
                            float* __restrict__ Z) {
    const int lane = threadIdx.x & 31;
    const int wave = threadIdx.x >> 5;
    const int tile = blockIdx.x * 8 + wave;          // 50000 tiles total
    const int row0 = (tile >> 4) * 16;               // 3125 row tiles
    const int col0 = (tile & 15) * 16;               // 16 col tiles (= HO/16)
    const int lm   = lane & 15;
    const int g    = lane >> 4;                      // K-pair group
    const int head = col0 >> 6;                      // 16-wide tiles never cross heads

    // A: contiguous K pair -> one b64 load per step
    const float* Ap = X + (size_t)(row0 + lm) * DIN_DIM + g * 2;
    // B: W layout (H, DIN, O): elem(h, k, o) at h*DIN*O + k*O + o
    const float* Bp = W + (size_t)head * (DIN_DIM * O_DIM)
                        + (size_t)(g * 2) * O_DIM + (col0 & 63) + lm;

    v8f acc = {};
    #pragma unroll 4
    for (int k0 = 0; k0 < DIN_DIM; k0 += 4) {
        v2f a = *(const v2f*)(Ap + k0);              // 8B aligned
        v2f b;
        b.x = Bp[(size_t)k0 * O_DIM];
        b.y = Bp[(size_t)k0 * O_DIM + O_DIM];
        acc = __builtin_amdgcn_wmma_f32_16x16x4_f32(
            /*neg_a=*/false, a, /*neg_b=*/false, b,
            /*c_mod=*/(short)0, acc, /*reuse_a=*/false, /*reuse_b=*/false);
    }

    float* Cp = Z + (size_t)row0 * HO + col0 + lm;
    #pragma unroll
    for (int r = 0; r < 8; ++r)
        Cp[(size_t)(r + g * 8) * HO] = acc[r];
}

// -------------------- per-node attention scores ------------------------------
// thread t -> node n = t>>2, head h = t&3 ; s = dot(z[n,h,:], a_{src,dst}[h,:])
__global__ void k_scores(const float* __restrict__ Z, const float* __restrict__ A,
                         float* __restrict__ ssrc, float* __restrict__ sdst) {
    int t = blockIdx.x * blockDim.x + threadIdx.x;
    if (t >= N_NODES * H_HEADS) return;
    int n = t >> 2, h = t & 3;
    const float4* zr = (const float4*)(Z + (size_t)n * HO + h * O_DIM);
    const float4* as = (const float4*)(A + h * 2 * O_DIM);
    const float4* ad = (const float4*)(A + h * 2 * O_DIM + O_DIM);
    float s0 = 0.0f, s1 = 0.0f;
    #pragma unroll
    for (int i = 0; i < O_DIM / 4; ++i) {
        float4 zv = zr[i], av = as[i], dv = ad[i];
        s0 += zv.x * av.x + zv.y * av.y + zv.z * av.z + zv.w * av.w;
        s1 += zv.x * dv.x + zv.y * dv.y + zv.z * dv.z + zv.w * dv.w;
    }
    ssrc[t] = s0;
    sdst[t] = s1;
}

// -------------------- pass 1: segment max over dst ---------------------------
__global__ void k_edge_max(const int* __restrict__ ei,
                           const float* __restrict__ ssrc,
                           const float* __restrict__ sdst,
                           unsigned* __restrict__ m) {
    int e = blockIdx.x * blockDim.x + threadIdx.x;
    if (e >= E_EDGES) return;
    int src = ei[e], dst = ei[E_EDGES + e];
    #pragma unroll
    for (int h = 0; h < H_HEADS; ++h) {
        float v = leaky(ssrc[src * H_HEADS + h] + sdst[dst * H_HEADS + h]);
        atomicMax(&m[dst * H_HEADS + h], encOrdered(v));
    }
}

// -------------------- pass 2: softmax denominator ----------------------------
__global__ void k_edge_expsum(const int* __restrict__ ei,
                              const float* __restrict__ ssrc,
                              const float* __restrict__ sdst,
                              const unsigned* __restrict__ m,
                              float* __restrict__ denom) {
    int e = blockIdx.x * blockDim.x + threadIdx.x;
    if (e >= E_EDGES) return;
    int src = ei[e], dst = ei[E_EDGES + e];
    #pragma unroll
    for (int h = 0; h < H_HEADS; ++h) {
        float v  = leaky(ssrc[src * H_HEADS + h] + sdst[dst * H_HEADS + h]);
        float mm = decOrdered(m[dst * H_HEADS + h]);
        atomicAdd(&denom[dst * H_HEADS + h], expf(v - mm));
    }
}

// -------------------- pass 3: weighted aggregation ---------------------------
// One wave32 per edge: 32 lanes x 8 iterations cover all 256 outputs with
// coalesced 128B atomic bursts; alpha recomputed from broadcast scalar loads.
__global__ void k_edge_aggregate(const int* __restrict__ ei,
                                 const float* __restrict__ ssrc,
                                 const float* __restrict__ sdst,
                                 const unsigned* __restrict__ m,
                                 const float* __restrict__ denom,
                                 const float* __restrict__ Z,
                                 float* __restrict__ out) {
    int gw   = (blockIdx.x * blockDim.x + threadIdx.x) >> 5;   // edge id
    int lane = threadIdx.x & 31;
    if (gw >= E_EDGES) return;
    int src = ei[gw], dst = ei[E_EDGES + gw];

    const float* zp = Z   + (size_t)src * HO;
    float*       op = out + (size_t)dst * HO;
    __builtin_prefetch(zp + lane * 8, 0, 0);   // kick off gather of z[src] row

    float alpha[H_HEADS];
    #pragma unroll
    for (int h = 0; h < H_HEADS; ++h) {
        float v  = leaky(ssrc[src * H_HEADS + h] + sdst[dst * H_HEADS + h]);
        float mm = decOrdered(m[dst * H_HEADS + h]);
        alpha[h] = expf(v - mm) / denom[dst * H_HEADS + h];
    }

    #pragma unroll
    for (int j = 0; j < HO / 32; ++j) {
        int idx = j * 32 + lane;
        atomicAdd(&op[idx], alpha[idx >> 6] * zp[idx]);
    }
}

// ------------------------------------------------------------------------------
extern "C" void kernel_launch(void* const* d_in, const int* in_sizes, int n_in,
                              void* d_out, int out_size, void* d_ws, size_t ws_size,
                              hipStream_t stream) {
    const float* features = (const float*)d_in[0];     // (N, 128)
    const float* W        = (const float*)d_in[1];     // (H, 128, 64)
    const float* A        = (const float*)d_in[2];     // (H, 128)
    const int*   ei       = (const int*)d_in[3];       // (2, E)
    float*       out      = (float*)d_out;             // (N, 256)

    // workspace layout
    float*    Z     = (float*)d_ws;                           // N*256 floats
    float*    ssrc  = Z + (size_t)N_NODES * HO;               // N*4
    float*    sdst  = ssrc + (size_t)N_NODES * H_HEADS;       // N*4
    float*    denom = sdst + (size_t)N_NODES * H_HEADS;       // N*4
    unsigned* m     = (unsigned*)(denom + (size_t)N_NODES * H_HEADS); // N*4

    const int TPB = 256;

    k_init<<<(N_NODES * HO + TPB - 1) / TPB, TPB, 0, stream>>>(out, denom, m);

    // 50000 16x16 tiles, 8 waves (tiles) per block -> 6250 blocks
    k_gemm_wmma<<<(N_NODES / 16) * (HO / 16) / 8, TPB, 0, stream>>>(features, W, Z);

    k_scores<<<(N_NODES * H_HEADS + TPB - 1) / TPB, TPB, 0, stream>>>(Z, A, ssrc, sdst);

    k_edge_max<<<(E_EDGES + TPB - 1) / TPB, TPB, 0, stream>>>(ei, ssrc, sdst, m);

    k_edge_expsum<<<(E_EDGES + TPB - 1) / TPB, TPB, 0, stream>>>(ei, ssrc, sdst, m, denom);

    // one wave per edge: E*32 threads
    k_edge_aggregate<<<(E_EDGES * 32 + TPB - 1) / TPB, TPB, 0, stream>>>(
        ei, ssrc, sdst, m, denom, Z, out);
}